// CNO_LReLu_738734375542
// MI455X (gfx1250) — compile-verified
//
#include <hip/hip_runtime.h>
#include <math.h>

// ---------------------------------------------------------------------------
// CNO block: antialiased bicubic up x2 -> LeakyReLU(0.01) -> antialiased
// bicubic down x2, along last axis of (32,256,2048) fp32.
//   out = W2 * leaky(W1 * x)   per row (8192 rows of length 2048)
// W1: 4096x2048, 5-tap band.  W2: 2048x4096, 9-tap band.
// Both stages done as 16x16 fp32 WMMA tiles (V_WMMA_F32_16X16X4_F32),
// fused through LDS so HBM traffic is just x (64MB) + out (64MB).
// ---------------------------------------------------------------------------

typedef float v2f __attribute__((ext_vector_type(2)));
typedef float v8f __attribute__((ext_vector_type(8)));

#define LIN   2048
#define LMID  4096
#define LOUT  2048
#define ROWS  8192          // B*C
#define WX    90            // LDS stride of x tile (>=88, even, low-conflict)
#define WY    162           // LDS stride of y tile (>=160, even, low-conflict)

#define W1_FLOATS (256 * 4 * 64)    // 256 blocks * 4 ksteps * 32 lanes * 2
#define W2_FLOATS (128 * 12 * 64)   // 128 blocks * 12 ksteps * 32 lanes * 2

// Keys cubic, a = -0.5 (torch/PIL antialias bicubic), double precision to
// match the numpy trace-time plan.
__device__ __forceinline__ double keys_cubic(double x) {
  double ax = fabs(x);
  if (ax <= 1.0) return (1.5 * ax - 2.5) * ax * ax + 1.0;
  if (ax < 2.0)  return -0.5 * (((ax - 5.0) * ax + 8.0) * ax - 4.0);
  return 0.0;
}

// ---------------------------------------------------------------------------
// Build W1^T fragments, pre-swizzled into WMMA B-matrix lane order.
// Fragment element (j, s, lane, v) holds W1[m = 16j + (lane&15)]
// evaluated at input column idx = (8j - 2) + k,  k = 4s + 2*(lane>>4) + v.
// ---------------------------------------------------------------------------
__global__ void cno_fill_w1(float* __restrict__ w1) {
  int tid = blockIdx.x * blockDim.x + threadIdx.x;
  if (tid >= 256 * 4 * 32) return;
  int j  = tid >> 7;
  int s  = (tid >> 5) & 3;
  int l  = tid & 31;
  int n  = l & 15;
  int hl = l >> 4;
  int m  = 16 * j + n;
  // upsample x2: scale=0.5, fscale=1, support=2, ntaps=5
  double c = ((double)m + 0.5) * 0.5;
  int ixmin = (int)floor(c - 1.5);
  double sum = 0.0;
  for (int t = 0; t < 5; ++t) {
    int idx = ixmin + t;
    if (idx >= 0 && idx < LIN) sum += keys_cubic((double)idx + 0.5 - c);
  }
  int b1 = 8 * j - 2;
  for (int v = 0; v < 2; ++v) {
    int k   = 4 * s + 2 * hl + v;
    int idx = b1 + k;
    int t   = idx - ixmin;
    double w = 0.0;
    if (t >= 0 && t < 5 && idx >= 0 && idx < LIN)
      w = keys_cubic((double)idx + 0.5 - c) / sum;
    w1[(size_t)((j * 4 + s) * 32 + l) * 2 + v] = (float)w;
  }
}

// ---------------------------------------------------------------------------
// Build W2^T fragments (downsample).  Fragment element (j, s, lane, v) holds
// W2[o = 16j + (lane&15)] at mid column idx = (32j - 8) + k.
// ---------------------------------------------------------------------------
__global__ void cno_fill_w2(float* __restrict__ w2) {
  int tid = blockIdx.x * blockDim.x + threadIdx.x;
  if (tid >= 128 * 12 * 32) return;
  int j  = tid / 384;
  int s  = (tid / 32) % 12;
  int l  = tid & 31;
  int n  = l & 15;
  int hl = l >> 4;
  int o  = 16 * j + n;
  // downsample x2: scale=2, fscale=2, support=4, ntaps=9
  double c = 2.0 * (double)o + 1.0;
  int ixmin = 2 * o - 3;                 // floor(c - 3.5)
  double sum = 0.0;
  for (int t = 0; t < 9; ++t) {
    int idx = ixmin + t;
    if (idx >= 0 && idx < LMID)
      sum += keys_cubic(((double)idx + 0.5 - c) * 0.5);
  }
  int b2 = 32 * j - 8;
  for (int v = 0; v < 2; ++v) {
    int k   = 4 * s + 2 * hl + v;
    int idx = b2 + k;
    int t   = idx - ixmin;
    double w = 0.0;
    if (t >= 0 && t < 9 && idx >= 0 && idx < LMID)
      w = keys_cubic(((double)idx + 0.5 - c) * 0.5) / sum;
    w2[(size_t)((j * 12 + s) * 32 + l) * 2 + v] = (float)w;
  }
}

// ---------------------------------------------------------------------------
// Main fused kernel: one wave32 per workgroup.
// Workgroup (q, rt): rows [16*rt, 16*rt+16), final outputs [64q, 64q+64).
//   x tile  : cols [64q-10, 64q+78)  -> LDS (clamped at edges)
//   y tile  : mids [128q-16, 128q+144) in LDS (leaky applied)
//   10 stage-1 WMMA tiles (K=16 each), 4 stage-2 WMMA tiles (K=40 each)
// ---------------------------------------------------------------------------
__global__ void __launch_bounds__(32)
cno_main(const float* __restrict__ x, const float* __restrict__ w1,
         const float* __restrict__ w2, float* __restrict__ out) {
  __shared__ __align__(16) float xs[16 * WX];
  __shared__ __align__(16) float ys[16 * WY];

  const int lane = threadIdx.x;       // 0..31
  const int q    = blockIdx.x;        // 0..31  (64-output slabs)
  const int r0   = blockIdx.y * 16;   // row tile base
  const int hl   = lane >> 4;
  const int ln   = lane & 15;

  // ---- stage x -> LDS (coalesced, clamped columns; clamped cols only ever
  //      meet zero weights so replicated values are harmless) ----
  const int xbase = 64 * q - 10;
  for (int row = 0; row < 16; ++row) {
    const float* xrow = x + (size_t)(r0 + row) * LIN;
    for (int c = lane; c < 88; c += 32) {
      int col = xbase + c;
      col = col < 0 ? 0 : (col > LIN - 1 ? LIN - 1 : col);
      xs[row * WX + c] = xrow[col];
    }
  }
  __syncthreads();

  // ---- stage 1: upsample, 10 mid-blocks jm = 8q-1 .. 8q+8 ----
  for (int jj = 0; jj < 10; ++jj) {
    int jm = 8 * q - 1 + jj;
    int jw = jm < 0 ? 0 : (jm > 255 ? 255 : jm);   // edge blocks: any finite
                                                   // weights ok, results hit
                                                   // zero W2 entries only.
    v8f acc = {0.f, 0.f, 0.f, 0.f, 0.f, 0.f, 0.f, 0.f};
    for (int s = 0; s < 4; ++s) {
      // A fragment: lane ln holds row ln, K = 4s + 2*hl + {0,1}
      v2f a = *(const v2f*)(xs + ln * WX + 8 * jj + 4 * s + 2 * hl);
      // B fragment: pre-swizzled weight pair
      v2f b = *(const v2f*)(w1 + (size_t)((jw * 4 + s) * 32 + lane) * 2);
      acc = __builtin_amdgcn_wmma_f32_16x16x4_f32(
          false, a, false, b, (short)0, acc, false, false);
    }
    // LeakyReLU on the D fragment, spill mid tile to LDS
    for (int g = 0; g < 8; ++g) {
      float v = acc[g];
      v = v >= 0.f ? v : 0.01f * v;
      ys[(g + 8 * hl) * WY + 16 * jj + ln] = v;
    }
  }
  __syncthreads();

  // ---- stage 2: downsample, 4 output blocks j = 4q + jo ----
  for (int jo = 0; jo < 4; ++jo) {
    int j = 4 * q + jo;
    v8f acc = {0.f, 0.f, 0.f, 0.f, 0.f, 0.f, 0.f, 0.f};
    // nonzero weight columns are i' in [5,43] -> K-steps 1..10 only
    for (int s = 1; s <= 10; ++s) {
      v2f a = *(const v2f*)(ys + ln * WY + 32 * jo + 8 + 4 * s + 2 * hl);
      v2f b = *(const v2f*)(w2 + (size_t)((j * 12 + s) * 32 + lane) * 2);
      acc = __builtin_amdgcn_wmma_f32_16x16x4_f32(
          false, a, false, b, (short)0, acc, false, false);
    }
    // store D fragment: per VGPR two contiguous 64B row segments
    for (int g = 0; g < 8; ++g) {
      out[(size_t)(r0 + g + 8 * hl) * LOUT + 16 * j + ln] = acc[g];
    }
  }
}

extern "C" void kernel_launch(void* const* d_in, const int* in_sizes, int n_in,
                              void* d_out, int out_size, void* d_ws, size_t ws_size,
                              hipStream_t stream) {
  const float* x = (const float*)d_in[0];
  float* out = (float*)d_out;
  float* w1  = (float*)d_ws;               // 256 KB
  float* w2  = (float*)d_ws + W1_FLOATS;   // 384 KB

  // Rebuild weight operators every call (deterministic, ~10us of work)
  cno_fill_w1<<<dim3((256 * 4 * 32 + 255) / 256), dim3(256), 0, stream>>>(w1);
  cno_fill_w2<<<dim3((128 * 12 * 32 + 255) / 256), dim3(256), 0, stream>>>(w2);

  dim3 grid(LOUT / 64, ROWS / 16);   // (32, 512)
  dim3 block(32);                    // one wave32
  cno_main<<<grid, block, 0, stream>>>(x, w1, w2, out);
}